// ViterbiViterbi_14594298871986
// MI455X (gfx1250) — compile-verified
//
#include <hip/hip_runtime.h>

// Viterbi-Viterbi phase recovery, 4 passes:
//  1) vv_winsum_angle : y = -z^4/|z|^4 (LDS, with halo), 50-tap sliding complex
//     sum done as banded 0/1 matmuls on V_WMMA_F32_16X16X4_F32, angle = atan2.
//  2) vv_wrap_partial : per-block sums of unwrap increments k_j in {-1,0,+1}.
//  3) vv_scan_offsets : exclusive scan of block partials (tiny).
//  4) vv_final        : per-strip scan -> K_j, out = z * e^{-i a/4} * i^{-K mod 4}.

typedef __attribute__((ext_vector_type(2))) float v2f;
typedef __attribute__((ext_vector_type(8))) float v8f;

#define HALF_W     25          // window 50, centered: j in [n-25, n+24]
#define TILE_BLOCK 2048        // outputs per block (8 waves x 16x16 tile)
#define HALO_LO    32
#define SPAN       (TILE_BLOCK + 32 + 40)   // 2120 y values per block
#define SCAN_BLOCK 16384
#define SCAN_STRIP 64
#define PI_F       3.14159265358979f

__device__ __forceinline__ int padIdx(int l) { return l + (l >> 4); }  // LDS bank de-conflict

__global__ __launch_bounds__(256) void vv_winsum_angle(
    const float* __restrict__ xr, const float* __restrict__ xi,
    float* __restrict__ ang, int N)
{
  __shared__ float2 ylds[SPAN + (SPAN >> 4) + 8];
  const int tid = threadIdx.x;
  const int B0  = blockIdx.x * TILE_BLOCK;

  // Fill y into LDS (zero-padded halo), y = -(z^4)/|z|^4 when |z|^4 > EPS.
  for (int l = tid; l < SPAN; l += 256) {
    int g = B0 - HALO_LO + l;
    float2 y = {0.f, 0.f};
    if (g >= 0 && g < N) {
      __builtin_prefetch(xr + g + 512, 0, 0);   // -> global_prefetch_b8
      __builtin_prefetch(xi + g + 512, 0, 0);
      float zr = xr[g], zi = xi[g];
      float z2r = zr * zr - zi * zi;
      float z2i = 2.f * zr * zi;
      float z4r = z2r * z2r - z2i * z2i;
      float z4i = 2.f * z2r * z2i;
      float r2  = zr * zr + zi * zi;
      float mag = r2 * r2;                      // |z^4|
      float inv = (mag > 1e-5f) ? (1.f / mag) : 1.f;
      y.x = -z4r * inv;                         // x^4 = (z*e^{i pi/4})^4 = -z^4
      y.y = -z4i * inv;
    }
    ylds[padIdx(l)] = y;
  }
  __syncthreads();

  // Each wave computes a 16x16 tile of windowed sums via banded-ones WMMA.
  const int wave  = tid >> 5;
  const int lane  = tid & 31;
  const int T     = B0 + wave * 256;            // tile origin: o[T + 16m + n]
  const int row   = lane & 15;                  // A-matrix row (M)
  const int nCol  = lane & 15;                  // B/D column (N)
  const int kBase = (lane < 16) ? 0 : 2;        // ISA: VGPR0=K0/K2, VGPR1=K1/K3

  v8f accRe = {0.f, 0.f, 0.f, 0.f, 0.f, 0.f, 0.f, 0.f};
  v8f accIm = {0.f, 0.f, 0.f, 0.f, 0.f, 0.f, 0.f, 0.f};

#pragma unroll
  for (int d = -2; d <= 2; ++d) {               // 5 shifted 16x16 y-blocks
#pragma unroll
    for (int q = 0; q < 4; ++q) {               // K=4 chunks
      // A[m][k] = y[T + 16*(m+d) + 4q + k], this lane supplies k = kBase, kBase+1
      int l0 = (T - B0 + HALO_LO) + 16 * (row + d) + 4 * q + kBase;
      float2 e0 = ylds[padIdx(l0)];
      float2 e1 = ylds[padIdx(l0 + 1)];
      v2f aRe = {e0.x, e1.x};
      v2f aIm = {e0.y, e1.y};
      // B[k][n] = 1 iff shift s = 16d+4q+k lies in window [n-25, n+24]
      int s0 = 16 * d + 4 * q + kBase;
      float b0 = (s0 >= nCol - HALF_W && s0 <= nCol + HALF_W - 1) ? 1.f : 0.f;
      float b1 = (s0 + 1 >= nCol - HALF_W && s0 + 1 <= nCol + HALF_W - 1) ? 1.f : 0.f;
      v2f bv = {b0, b1};
      accRe = __builtin_amdgcn_wmma_f32_16x16x4_f32(false, aRe, false, bv,
                                                    (short)0, accRe, false, false);
      accIm = __builtin_amdgcn_wmma_f32_16x16x4_f32(false, aIm, false, bv,
                                                    (short)0, accIm, false, false);
    }
  }

  // D layout: lanes 0-15 -> M=r, lanes 16-31 -> M=8+r, N = lane&15.
  const int mBase = (lane < 16) ? 0 : 8;
#pragma unroll
  for (int r = 0; r < 8; ++r) {
    int idx = T + 16 * (mBase + r) + nCol;
    if (idx < N) ang[idx] = atan2f(accIm[r], accRe[r]);
  }
}

__global__ __launch_bounds__(256) void vv_wrap_partial(
    const float* __restrict__ ang, int* __restrict__ partial, int N)
{
  __shared__ int sh[256];
  int tid  = threadIdx.x;
  int base = blockIdx.x * SCAN_BLOCK + tid * SCAN_STRIP;
  int s = 0;
  if (base < N) {
    int   j0   = base > 0 ? base : 1;           // k_0 := 0
    int   jEnd = min(base + SCAN_STRIP, N);
    float prev = ang[j0 - 1];
    for (int j = j0; j < jEnd; ++j) {
      float a = ang[j];
      float d = a - prev;
      if (d > PI_F) s -= 1; else if (d < -PI_F) s += 1;
      prev = a;
    }
  }
  sh[tid] = s;
  __syncthreads();
  for (int off = 128; off > 0; off >>= 1) {
    if (tid < off) sh[tid] += sh[tid + off];
    __syncthreads();
  }
  if (tid == 0) partial[blockIdx.x] = sh[0];
}

__global__ void vv_scan_offsets(const int* __restrict__ partial,
                                int* __restrict__ blockOff, int nb)
{
  if (threadIdx.x == 0 && blockIdx.x == 0) {
    int run = 0;
    for (int b = 0; b < nb; ++b) { blockOff[b] = run; run += partial[b]; }
  }
}

__global__ __launch_bounds__(256) void vv_final(
    const float* __restrict__ xr, const float* __restrict__ xi,
    const float* __restrict__ ang, const int* __restrict__ blockOff,
    float* __restrict__ out, int N)
{
  __shared__ int sh[256];
  __shared__ int shExcl[256];
  int tid  = threadIdx.x;
  int base = blockIdx.x * SCAN_BLOCK + tid * SCAN_STRIP;

  // Pass 1: per-strip wrap sums.
  int s = 0;
  if (base < N) {
    int   j0   = base > 0 ? base : 1;
    int   jEnd = min(base + SCAN_STRIP, N);
    float prev = ang[j0 - 1];
    for (int j = j0; j < jEnd; ++j) {
      float a = ang[j];
      float d = a - prev;
      if (d > PI_F) s -= 1; else if (d < -PI_F) s += 1;
      prev = a;
    }
  }
  sh[tid] = s;
  __syncthreads();
  if (tid == 0) {                               // 256-entry exclusive scan
    int run = 0;
    for (int t = 0; t < 256; ++t) { shExcl[t] = run; run += sh[t]; }
  }
  __syncthreads();
  if (base >= N) return;

  // Pass 2: K_j = total wraps through j; emit z * e^{-i a/4} * e^{-i (pi/2) K}.
  int   K    = blockOff[blockIdx.x] + shExcl[tid];   // = K_{base-1}
  int   jEnd = min(base + SCAN_STRIP, N);
  float prev = (base > 0) ? ang[base - 1] : 0.f;
  for (int j = base; j < jEnd; ++j) {
    __builtin_prefetch(xr + j + 64, 0, 0);
    float a = ang[j];
    if (j > 0) {
      float d = a - prev;
      if (d > PI_F) K -= 1; else if (d < -PI_F) K += 1;
    }
    prev = a;
    float zr = xr[j], zi = xi[j];
    float sn, cs;
    __sincosf(a * 0.25f, &sn, &cs);
    float re = zr * cs + zi * sn;               // z * (cos - i sin)
    float im = zi * cs - zr * sn;
    int q = K & 3;                              // e^{-i pi/2 * (K mod 4)}
    float orr, oii;
    if      (q == 0) { orr =  re; oii =  im; }
    else if (q == 1) { orr =  im; oii = -re; }  // * -i
    else if (q == 2) { orr = -re; oii = -im; }  // * -1
    else             { orr = -im; oii =  re; }  // * +i
    out[2 * j]     = orr;
    out[2 * j + 1] = oii;
  }
}

extern "C" void kernel_launch(void* const* d_in, const int* in_sizes, int n_in,
                              void* d_out, int out_size, void* d_ws, size_t ws_size,
                              hipStream_t stream)
{
  (void)n_in; (void)out_size; (void)ws_size;
  const float* xr = (const float*)d_in[0];
  const float* xi = (const float*)d_in[1];
  float* out = (float*)d_out;
  const int N = in_sizes[0];

  // Workspace: [ angles: N floats ][ partials: nb ints ][ offsets: nb ints ]
  float* ang      = (float*)d_ws;
  int    nb       = (N + SCAN_BLOCK - 1) / SCAN_BLOCK;
  int*   partial  = (int*)((char*)d_ws + (size_t)N * sizeof(float));
  int*   blockOff = partial + nb;

  int gA = (N + TILE_BLOCK - 1) / TILE_BLOCK;
  vv_winsum_angle<<<gA, 256, 0, stream>>>(xr, xi, ang, N);
  vv_wrap_partial<<<nb, 256, 0, stream>>>(ang, partial, N);
  vv_scan_offsets<<<1, 1, 0, stream>>>(partial, blockOff, nb);
  vv_final<<<nb, 256, 0, stream>>>(xr, xi, ang, blockOff, out, N);
}